// NewsImpactModel_14680198218193
// MI455X (gfx1250) — compile-verified
//
#include <hip/hip_runtime.h>
#include <hip/hip_bf16.h>
#include <math.h>

#define NDAYS 4096
#define NNEWS 8192
#define PDF_PAD 16
#define NWAVES 8   // waves per block in conv kernel (K-split factor)

typedef __attribute__((ext_vector_type(2))) float v2f;
typedef __attribute__((ext_vector_type(8))) float v8f;

__device__ __forceinline__ float softplusf(float x) {
  return (x > 20.0f) ? x : log1pf(expf(x));
}

// ---------------------------------------------------------------------------
// K1: build gamma pdf tables pdf_h[d], d=0..4095, for the 3 horizons.
// ---------------------------------------------------------------------------
__global__ void build_pdf_kernel(
    const float* rk_s, const float* rt_s, const float* rl_s,
    const float* rk_m, const float* rt_m, const float* rl_m,
    const float* rk_l, const float* rt_l, const float* rl_l,
    float* __restrict__ pdf) {
  int h = blockIdx.x;
  float rk = (h == 0) ? rk_s[0] : (h == 1) ? rk_m[0] : rk_l[0];
  float rt = (h == 0) ? rt_s[0] : (h == 1) ? rt_m[0] : rt_l[0];
  float rl = (h == 0) ? rl_s[0] : (h == 1) ? rl_m[0] : rl_l[0];
  float k     = softplusf(rk) + 1.0f + 1e-6f;
  float theta = softplusf(rt) + 1e-6f;
  float loc   = softplusf(rl);
  float rate  = 1.0f / theta;
  float lg    = lgammaf(k);
  float klogr = k * logf(rate);
  for (int d = threadIdx.x; d < NDAYS; d += blockDim.x) {
    float xs = (float)d - loc;
    float xc = fmaxf(xs, 1e-9f);
    float lp = klogr + (k - 1.0f) * logf(xc) - rate * xc - lg;
    pdf[h * NDAYS + d] = (xs > 0.0f) ? expf(lp) : 0.0f;
  }
}

// ---------------------------------------------------------------------------
// K2: inclusive prefix-max of each pdf table (peak[n] = pmax[4095 - day_idx]).
// ---------------------------------------------------------------------------
__global__ void prefix_max_kernel(const float* __restrict__ pdf,
                                  float* __restrict__ pmax) {
  __shared__ float cm[1024];
  int h = blockIdx.x;
  int tid = threadIdx.x;  // 1024 threads, 4 elements each
  const float* P = pdf + h * NDAYS;
  float la[4];
  float run = -INFINITY;
#pragma unroll
  for (int i = 0; i < 4; ++i) {
    run = fmaxf(run, P[4 * tid + i]);
    la[i] = run;
  }
  cm[tid] = run;
  __syncthreads();
  for (int off = 1; off < 1024; off <<= 1) {
    float v = cm[tid];
    if (tid >= off) v = fmaxf(v, cm[tid - off]);
    __syncthreads();
    cm[tid] = v;
    __syncthreads();
  }
  float ex = (tid > 0) ? cm[tid - 1] : -INFINITY;
#pragma unroll
  for (int i = 0; i < 4; ++i)
    pmax[h * NDAYS + 4 * tid + i] = fmaxf(ex, la[i]);
}

// ---------------------------------------------------------------------------
// K3a/K3b: zero per-day coefficient buckets, then scatter news into them:
//   c_h[day] += relevance * pct_h / (peak + 1e-9)
// ---------------------------------------------------------------------------
__global__ void zero_kernel(float* __restrict__ c) {
  int i = blockIdx.x * blockDim.x + threadIdx.x;
  if (i < 3 * NDAYS) c[i] = 0.0f;
}

__global__ void scatter_kernel(const int* __restrict__ day_index,
                               const float* __restrict__ pct_s,
                               const float* __restrict__ pct_m,
                               const float* __restrict__ pct_l,
                               const float* __restrict__ relevance,
                               const float* __restrict__ pmax,
                               float* __restrict__ c) {
  int n = blockIdx.x * blockDim.x + threadIdx.x;
  if (n >= NNEWS) return;
  int di = day_index[n];
  float rel = relevance[n];
  int pidx = (NDAYS - 1) - di;
  float pcts[3] = {pct_s[n], pct_m[n], pct_l[n]};
#pragma unroll
  for (int h = 0; h < 3; ++h) {
    float pk = pmax[h * NDAYS + pidx] + 1e-9f;
    atomicAdd(&c[h * NDAYS + di], rel * pcts[h] / pk);
  }
}

// ---------------------------------------------------------------------------
// K4: causal convolution out[t] = sum_h sum_{d0<=t} c_h[d0]*pdf_h[t-d0]
// via fp32 WMMA (lower-triangular Toeplitz matvec, exact fp32 precision).
// One BLOCK per 16-day output tile; the causal K-range is split across the
// 8 waves of the block (wave s takes d0 = 4s, 4s+32, ...) so the critical
// path is ~8x shorter than one-wave-per-tile; partial D tiles are combined
// with a small LDS reduction.
// fp32 A layout: lanes 0-15 hold K=0 (v0),K=1 (v1); lanes 16-31 K=2,K=3.
// D column 0 lives in lanes 0 (rows 0-7) and 16 (rows 8-15).
// ---------------------------------------------------------------------------
__global__ __launch_bounds__(256) void conv_wmma_kernel(
    const float* __restrict__ pdf, const float* __restrict__ c,
    float* __restrict__ out) {
  __shared__ float spdf[3 * (NDAYS + PDF_PAD)];  // ~48 KB, zero-padded heads
  __shared__ float partial[NWAVES][16];
  int tid = threadIdx.x;
  for (int i = tid; i < 3 * (NDAYS + PDF_PAD); i += 256) {
    int h = i / (NDAYS + PDF_PAD);
    int j = i - h * (NDAYS + PDF_PAD);
    spdf[i] = (j < PDF_PAD) ? 0.0f : pdf[h * NDAYS + (j - PDF_PAD)];
  }
  __syncthreads();

  int lane = tid & 31;
  int wave = tid >> 5;       // K-split index: 0..7
  int t0 = blockIdx.x * 16;  // this block's output tile
  int m   = lane & 15;       // A row / B column index
  int grp = lane >> 4;       // 0: K={0,1}, 1: K={2,3}
  int k0  = grp * 2;

  v8f acc = {0.0f, 0.0f, 0.0f, 0.0f, 0.0f, 0.0f, 0.0f, 0.0f};
  for (int h = 0; h < 3; ++h) {
    const float* P = &spdf[h * (NDAYS + PDF_PAD) + PDF_PAD];
    const float* C = &c[h * NDAYS];
    for (int d0 = wave * 4; d0 < t0 + 16; d0 += 4 * NWAVES) {
      int ia = t0 + m - d0 - k0;  // lag for K=k0 (a.x); a.y is lag-1 (K=k0+1)
      v2f a;
      a.x = P[ia];
      a.y = P[ia - 1];
      float b0 = C[d0 + k0];
      float b1 = C[d0 + k0 + 1];
      v2f b;
      b.x = (m == 0) ? b0 : 0.0f;  // only column n==0 carries the vector
      b.y = (m == 0) ? b1 : 0.0f;
      acc = __builtin_amdgcn_wmma_f32_16x16x4_f32(
          false, a, false, b, (short)0, acc, false, false);
    }
  }
  if (m == 0) {  // lanes 0 and 16 hold D column 0 (rows 0-7 / 8-15)
#pragma unroll
    for (int v = 0; v < 8; ++v) partial[wave][grp * 8 + v] = acc[v];
  }
  __syncthreads();
  if (tid < 16) {
    float s = 0.0f;
#pragma unroll
    for (int w = 0; w < NWAVES; ++w) s += partial[w][tid];
    out[t0 + tid] = s;
  }
}

// ---------------------------------------------------------------------------
extern "C" void kernel_launch(void* const* d_in, const int* in_sizes, int n_in,
                              void* d_out, int out_size, void* d_ws, size_t ws_size,
                              hipStream_t stream) {
  (void)in_sizes; (void)n_in; (void)out_size; (void)ws_size;
  // d_in order: 0 pred_time (unused: arange), 1 day_index(i32), 2..4 pct_s/m/l,
  // 5 relevance, 6..14 raw scalars (k,theta,loc) x (short,medium,long)
  const int*   day_index = (const int*)d_in[1];
  const float* pct_s     = (const float*)d_in[2];
  const float* pct_m     = (const float*)d_in[3];
  const float* pct_l     = (const float*)d_in[4];
  const float* relevance = (const float*)d_in[5];

  float* ws   = (float*)d_ws;
  float* pdf  = ws;              // 3*4096 floats
  float* pmax = ws + 3 * NDAYS;  // 3*4096 floats
  float* cbuf = ws + 6 * NDAYS;  // 3*4096 floats

  build_pdf_kernel<<<3, 256, 0, stream>>>(
      (const float*)d_in[6],  (const float*)d_in[7],  (const float*)d_in[8],
      (const float*)d_in[9],  (const float*)d_in[10], (const float*)d_in[11],
      (const float*)d_in[12], (const float*)d_in[13], (const float*)d_in[14],
      pdf);
  prefix_max_kernel<<<3, 1024, 0, stream>>>(pdf, pmax);
  zero_kernel<<<(3 * NDAYS + 255) / 256, 256, 0, stream>>>(cbuf);
  scatter_kernel<<<(NNEWS + 255) / 256, 256, 0, stream>>>(
      day_index, pct_s, pct_m, pct_l, relevance, pmax, cbuf);
  conv_wmma_kernel<<<NDAYS / 16, 256, 0, stream>>>(pdf, cbuf, (float*)d_out);
}